// PureTriadicBrain_39324720562582
// MI455X (gfx1250) — compile-verified
//
#include <hip/hip_runtime.h>
#include <hip/hip_bf16.h>
#include <math.h>

// ---------------- sizes ----------------
#define BB 16
#define NN 128
#define DD 64
#define BN (BB*NN)            // 2048
#define BND (BN*DD)           // 131072
#define BNN (BB*NN*NN)        // 262144
#define BNDD ((size_t)BN*DD*DD) // 8388608

typedef __attribute__((ext_vector_type(2))) float v2f;
typedef __attribute__((ext_vector_type(8))) float v8f;

// =====================================================================
// K1: per-(b,n) GEMVs:  q = state@W3, k = state@W2, raw_pred = state@W1
//     prediction = tanh(raw - tanh(raw)*0.6) * sqrt(2+sqrt(2))
// =====================================================================
__global__ void k1_qk_pred(const float* __restrict__ state,
                           const float* __restrict__ W1,
                           const float* __restrict__ W2,
                           const float* __restrict__ W3,
                           float* __restrict__ q, float* __restrict__ kv,
                           float* __restrict__ pred) {
  int bn = blockIdx.x;          // 0..2047
  int t  = threadIdx.x;         // 0..63 (output column)
  __shared__ float s[DD];
  s[t] = state[(size_t)bn*DD + t];
  __syncthreads();
  const float* w1 = W1 + (size_t)bn*DD*DD;
  const float* w2 = W2 + (size_t)bn*DD*DD;
  const float* w3 = W3 + (size_t)bn*DD*DD;
  float a1 = 0.f, a2 = 0.f, a3 = 0.f;
#pragma unroll 4
  for (int j = 0; j < DD; ++j) {
    float sj = s[j];
    a1 = fmaf(sj, w1[j*DD + t], a1);
    a2 = fmaf(sj, w2[j*DD + t], a2);
    a3 = fmaf(sj, w3[j*DD + t], a3);
  }
  q [(size_t)bn*DD + t] = a3;   // q uses W3
  kv[(size_t)bn*DD + t] = a2;   // k uses W2
  float raw = a1 - tanhf(a1)*0.6f;
  pred[(size_t)bn*DD + t] = tanhf(raw) * 1.8477590650225735f; // sqrt(2+sqrt(2))
}

// =====================================================================
// K2: raw_A[b] = q[b] @ k[b]^T / sqrt(D) + noise_A*1e-5  via f32 WMMA
//     one wave per 16-row strip; 8 column tiles; K-loop of 16 wmma ops
// =====================================================================
__global__ __launch_bounds__(32)
void k2_scores_wmma(const float* __restrict__ q, const float* __restrict__ kv,
                    const float* __restrict__ noiseA, float* __restrict__ rawA) {
  int bid  = blockIdx.x;        // B*8 = 128
  int b    = bid >> 3;
  int m0   = (bid & 7) * 16;    // row strip
  int lane = threadIdx.x;       // 0..31
  int lm   = lane & 15, hi = lane >> 4;
  const float* qb = q  + (size_t)b*NN*DD;
  const float* kb = kv + (size_t)b*NN*DD;
  v8f acc[8];
  v8f z = {0.f,0.f,0.f,0.f,0.f,0.f,0.f,0.f};
#pragma unroll
  for (int t = 0; t < 8; ++t) acc[t] = z;
  for (int k0 = 0; k0 < DD; k0 += 4) {
    v2f a;                                        // A: 16x4 tile of q
    a.x = qb[(m0 + lm)*DD + k0 + 2*hi];
    a.y = qb[(m0 + lm)*DD + k0 + 2*hi + 1];
#pragma unroll
    for (int t = 0; t < 8; ++t) {
      int n0 = t * 16;
      v2f bf;                                     // B: 4x16 tile of k^T
      bf.x = kb[(n0 + lm)*DD + k0 + 2*hi];
      bf.y = kb[(n0 + lm)*DD + k0 + 2*hi + 1];
      acc[t] = __builtin_amdgcn_wmma_f32_16x16x4_f32(
          false, a, false, bf, (short)0, acc[t], false, false);
    }
  }
#pragma unroll
  for (int t = 0; t < 8; ++t) {
    int n0 = t * 16;
#pragma unroll
    for (int r = 0; r < 8; ++r) {
      int m = m0 + r + 8*hi;                      // C layout: vgpr r -> M=r / M=r+8
      int n = n0 + lm;
      size_t idx = ((size_t)b*NN + m)*NN + n;
      rawA[idx] = acc[t][r]*0.125f + noiseA[idx]*1e-5f;
    }
  }
}

// =====================================================================
// K3: row softmax -> keep mask -> re-softmax -> A EMA, zero rows 0,1
// =====================================================================
__global__ void k3_route(const float* __restrict__ rawA, const float* __restrict__ A,
                         float* __restrict__ Anew) {
  int bn = blockIdx.x;              // 2048 rows
  int n  = bn & (NN-1);
  int m  = threadIdx.x;             // 0..127 column
  __shared__ float red[NN];
  size_t idx = (size_t)bn*NN + m;
  float x = rawA[idx];
  red[m] = x; __syncthreads();
  for (int s = 64; s > 0; s >>= 1) { if (m < s) red[m] = fmaxf(red[m], red[m+s]); __syncthreads(); }
  float mx = red[0]; __syncthreads();
  float e = __expf(x - mx);
  red[m] = e; __syncthreads();
  for (int s = 64; s > 0; s >>= 1) { if (m < s) red[m] += red[m+s]; __syncthreads(); }
  float sm = red[0]; __syncthreads();
  float pd = e / sm;
  bool keep = (pd > (1.0f/129.0f)) || (x == mx);
  red[m] = keep ? e : 0.0f; __syncthreads();
  for (int s = 64; s > 0; s >>= 1) { if (m < s) red[m] += red[m+s]; __syncthreads(); }
  float sm2 = red[0];
  float P = keep ? (e / sm2) : 0.0f;
  float an = A[idx]*0.99f + P*0.01f;
  if (n < 2) an = 0.0f;
  Anew[idx] = an;
}

// =====================================================================
// K4: target[b] = A_new[b] @ output[b]  via f32 WMMA; clamp rows 0,1
// =====================================================================
__global__ __launch_bounds__(32)
void k4_target_wmma(const float* __restrict__ Anew, const float* __restrict__ outp,
                    const float* __restrict__ eye, const float* __restrict__ stomach,
                    float* __restrict__ target) {
  int bid  = blockIdx.x;        // B*8 = 128
  int b    = bid >> 3;
  int m0   = (bid & 7) * 16;    // n-row strip
  int lane = threadIdx.x;
  int lm   = lane & 15, hi = lane >> 4;
  const float* Ab = Anew + (size_t)b*NN*NN;
  const float* Ob = outp + (size_t)b*NN*DD;
  v8f acc[4];
  v8f z = {0.f,0.f,0.f,0.f,0.f,0.f,0.f,0.f};
#pragma unroll
  for (int t = 0; t < 4; ++t) acc[t] = z;
  for (int k0 = 0; k0 < NN; k0 += 4) {
    v2f a;
    a.x = Ab[(m0 + lm)*NN + k0 + 2*hi];
    a.y = Ab[(m0 + lm)*NN + k0 + 2*hi + 1];
#pragma unroll
    for (int t = 0; t < 4; ++t) {
      int d0 = t * 16;
      v2f bf;
      bf.x = Ob[(k0 + 2*hi    )*DD + d0 + lm];
      bf.y = Ob[(k0 + 2*hi + 1)*DD + d0 + lm];
      acc[t] = __builtin_amdgcn_wmma_f32_16x16x4_f32(
          false, a, false, bf, (short)0, acc[t], false, false);
    }
  }
#pragma unroll
  for (int t = 0; t < 4; ++t) {
    int d0 = t * 16;
#pragma unroll
    for (int r = 0; r < 8; ++r) {
      int m = m0 + r + 8*hi;
      int d = d0 + lm;
      float v = acc[t][r];
      if (m == 0)      v = eye[b*DD + d];
      else if (m == 1) v = stomach[b*DD + d];
      target[((size_t)b*NN + m)*DD + d] = v;
    }
  }
}

// =====================================================================
// K5a: deterministic partial reductions: err sum/sumsq, per-b sum|state|
// =====================================================================
__global__ void k5a_stats(const float* __restrict__ pred, const float* __restrict__ target,
                          const float* __restrict__ state,
                          float* __restrict__ part, float* __restrict__ invmag) {
  __shared__ float r1[256], r2[256];
  int t = threadIdx.x;
  if (blockIdx.x < 64) {
    int base = blockIdx.x * (BND/64);               // 2048 elems per block
    float s = 0.f, sq = 0.f;
    for (int i = t; i < BND/64; i += 256) {
      float e = pred[base+i] - target[base+i];
      s += e; sq += e*e;
    }
    r1[t] = s; r2[t] = sq; __syncthreads();
    for (int k = 128; k > 0; k >>= 1) { if (t < k) { r1[t]+=r1[t+k]; r2[t]+=r2[t+k]; } __syncthreads(); }
    if (t == 0) { part[blockIdx.x*2] = r1[0]; part[blockIdx.x*2+1] = r2[0]; }
  } else {
    int b = blockIdx.x - 64;
    const float* sb = state + (size_t)b*NN*DD;
    float s = 0.f;
    for (int i = t; i < NN*DD; i += 256) s += fabsf(sb[i]);
    r1[t] = s; __syncthreads();
    for (int k = 128; k > 0; k >>= 1) { if (t < k) r1[t] += r1[t+k]; __syncthreads(); }
    if (t == 0) invmag[b] = 1.0f / (1.0f + r1[0]);
  }
}

__global__ void k5b_std(const float* __restrict__ part, float* __restrict__ n2s) {
  __shared__ float r1[64], r2[64];
  int t = threadIdx.x;      // 64
  r1[t] = part[2*t]; r2[t] = part[2*t+1]; __syncthreads();
  for (int k = 32; k > 0; k >>= 1) { if (t < k) { r1[t]+=r1[t+k]; r2[t]+=r2[t+k]; } __syncthreads(); }
  if (t == 0) {
    const float M = (float)BND;
    float var = (r2[0] - r1[0]*r1[0]/M) / (M - 1.0f);   // ddof=1
    n2s[0] = sqrtf(fmaxf(var, 0.0f)) * 0.8f;            // NOISE_SCALE
  }
}

// =====================================================================
// K6: fused per-(b,n) plasticity update -> W1n/W2n/W3n
// =====================================================================
__global__ void k6_update(const float* __restrict__ pred, const float* __restrict__ target,
                          const float* __restrict__ state, const float* __restrict__ Ebase,
                          const float* __restrict__ n1raw, const float* __restrict__ n2raw,
                          const float* __restrict__ W1, const float* __restrict__ W2,
                          const float* __restrict__ W3, const float* __restrict__ M1,
                          const float* __restrict__ M2, const float* __restrict__ M3,
                          const float* __restrict__ invmag, const float* __restrict__ n2s,
                          const int* __restrict__ stepc,
                          float* __restrict__ W1n, float* __restrict__ W2n,
                          float* __restrict__ W3n) {
  int bn = blockIdx.x;                 // 2048
  int b  = bn >> 7;
  int j  = threadIdx.x;                // 0..63 column
  __shared__ float errS[DD], pmS[DD];
  __shared__ float red1[DD], red2[DD], red3[DD];
  size_t ro = (size_t)bn*DD;
  float e = pred[ro+j] - target[ro+j];

  // E_curr = softmax(err) over D
  red1[j] = e; __syncthreads();
  for (int k = 32; k > 0; k >>= 1) { if (j < k) red1[j] = fmaxf(red1[j], red1[j+k]); __syncthreads(); }
  float mx = red1[0]; __syncthreads();
  float ex = __expf(e - mx);
  red1[j] = ex; __syncthreads();
  for (int k = 32; k > 0; k >>= 1) { if (j < k) red1[j] += red1[j+k]; __syncthreads(); }
  float Ec = ex / red1[0]; __syncthreads();

  float Eb0 = Ebase[ro+j];
  float Eb  = (Eb0 == 0.0f) ? Ec : Eb0;          // bmask substitution
  Eb = Eb*0.95f + 0.05f*Ec;                      // EMA_SPEED
  float adv   = Ec - Eb;
  float R     = -(adv * rsqrtf(adv*adv + 1e-6f)); // rms_norm over size-1 axis
  float plast = 1.0f - R;
  float maskv = (Eb > 0.0f) ? 1.0f : 0.0f;
  float pm    = plast * maskv;                   // per-row (i) factor

  float ns = state[ro+j] + n1raw[ro+j]*invmag[b] + n2raw[ro+j]*n2s[0];
  errS[j] = e; pmS[j] = pm;

  // rms of lg = -outer(err, ns):  mean(lg^2) = (sum e^2)(sum ns^2)/4096
  red1[j] = e*e; red2[j] = ns*ns; __syncthreads();
  for (int k = 32; k > 0; k >>= 1) { if (j < k) { red1[j]+=red1[j+k]; red2[j]+=red2[j+k]; } __syncthreads(); }
  float invlg = rsqrtf(red1[0]*red2[0]*(1.0f/4096.0f) + 1e-6f);
  __syncthreads();

  // travelling gaussian column mask (column j)
  float sc = (float)stepc[0] + 1.0f;
  float center = fmodf(sc*0.5f, 64.0f);
  float diff = fabsf((float)j - center);
  diff = fminf(diff, 64.0f - diff);
  float g = __expf(-diff*diff / 0.020001f);      // 2*WIDTH^2+1e-6

  size_t wo = (size_t)bn*DD*DD;
  for (int i = 0; i < DD; ++i) {
    float lg    = -errS[i]*ns*invlg;             // lg[i][j], ns is this column's value
    float hgrad = pmS[i]*lg;
    size_t o = wo + (size_t)i*DD + j;
    float w1 = W1[o], w2 = W2[o], w3 = W3[o];
    float g1 = hgrad + 0.01f*w3 - 0.01f*w1;      // LAT_DECAY / WD
    float g2 = hgrad + 0.01f*w1 - 0.01f*w2;
    float g3 = hgrad + 0.01f*w2 - 0.01f*w3;
    float m1 = M1[o]*0.4f + 0.6f*(g1*g);         // MOMENTUM
    float m2 = M2[o]*0.4f + 0.6f*(g2*g);
    float m3 = M3[o]*0.4f + 0.6f*(g3*g);
    float t1 = w1 + 0.0033f*m1;                  // LR
    float t2 = w2 + 0.0033f*m2;
    float t3 = w3 + 0.0033f*m3;
    red1[j] = t1*t1; red2[j] = t2*t2; red3[j] = t3*t3; __syncthreads();
    for (int k = 32; k > 0; k >>= 1) {
      if (j < k) { red1[j]+=red1[j+k]; red2[j]+=red2[j+k]; red3[j]+=red3[j+k]; }
      __syncthreads();
    }
    float i1 = rsqrtf(red1[0]*(1.0f/64.0f) + 1e-6f);  // row rms_norm
    float i2 = rsqrtf(red2[0]*(1.0f/64.0f) + 1e-6f);
    float i3 = rsqrtf(red3[0]*(1.0f/64.0f) + 1e-6f);
    __syncthreads();
    W1n[o] = t1*i1; W2n[o] = t2*i2; W3n[o] = t3*i3;   // W_SCALE = 1
  }
}

// =====================================================================
extern "C" void kernel_launch(void* const* d_in, const int* in_sizes, int n_in,
                              void* d_out, int out_size, void* d_ws, size_t ws_size,
                              hipStream_t stream) {
  const float* eye     = (const float*)d_in[0];
  const float* stomach = (const float*)d_in[1];
  const float* state   = (const float*)d_in[2];
  const float* output  = (const float*)d_in[3];
  const float* W1      = (const float*)d_in[4];
  const float* W2      = (const float*)d_in[5];
  const float* W3      = (const float*)d_in[6];
  const float* M1      = (const float*)d_in[7];
  const float* M2      = (const float*)d_in[8];
  const float* M3      = (const float*)d_in[9];
  const float* Ebase   = (const float*)d_in[10];
  const float* A       = (const float*)d_in[11];
  const float* noiseA  = (const float*)d_in[12];
  const float* n1raw   = (const float*)d_in[13];
  const float* n2raw   = (const float*)d_in[14];
  const int*   stepc   = (const int*)d_in[15];

  float* out    = (float*)d_out;
  float* pred   = out;                        // (B,N,D)
  float* target = out + BND;                  // (B,N,D)
  float* Anew   = out + 2*BND;                // (B,N,N)
  float* W1n    = out + 2*BND + BNN;          // (B,N,D,D)
  float* W2n    = W1n + BNDD;
  float* W3n    = W2n + BNDD;

  float* ws     = (float*)d_ws;
  float* q      = ws;                         // (B,N,D)
  float* kv     = ws + BND;                   // (B,N,D)
  float* rawA   = ws + 2*BND;                 // (B,N,N)
  float* part   = ws + 2*BND + BNN;           // 128 partials
  float* invmag = part + 128;                 // 16
  float* n2s    = invmag + 16;                // 1

  k1_qk_pred   <<<BN, DD, 0, stream>>>(state, W1, W2, W3, q, kv, pred);
  k2_scores_wmma<<<BB*8, 32, 0, stream>>>(q, kv, noiseA, rawA);
  k3_route     <<<BN, NN, 0, stream>>>(rawA, A, Anew);
  k4_target_wmma<<<BB*8, 32, 0, stream>>>(Anew, output, eye, stomach, target);
  k5a_stats    <<<80, 256, 0, stream>>>(pred, target, state, part, invmag);
  k5b_std      <<<1, 64, 0, stream>>>(part, n2s);
  k6_update    <<<BN, DD, 0, stream>>>(pred, target, state, Ebase, n1raw, n2raw,
                                       W1, W2, W3, M1, M2, M3,
                                       invmag, n2s, stepc, W1n, W2n, W3n);
}